// GINConv_22531398435299
// MI455X (gfx1250) — compile-verified
//
#include <hip/hip_runtime.h>

#define N_NODES 100000
#define N_EDGES 1600000
#define IN_DIM  32
#define OUT_DIM 32
#define N_TILES (N_NODES / 16)   // 6250, exact: every 16-row tile is full

typedef __attribute__((ext_vector_type(2))) float v2f;
typedef __attribute__((ext_vector_type(8))) float v8f;

// ---------------------------------------------------------------------------
// Kernel 1: zero the aggregation buffer (d_ws is poisoned and we accumulate
// into it -> must clear on every launch for determinism).
// ---------------------------------------------------------------------------
__global__ __launch_bounds__(256) void gin_zero_kernel(float4* __restrict__ agg4, int n4) {
    int i = blockIdx.x * 256 + threadIdx.x;
    if (i < n4) agg4[i] = float4{0.f, 0.f, 0.f, 0.f};
}

// ---------------------------------------------------------------------------
// Kernel 2: edge-parallel scatter-add. Each lane handles 2 dims of an edge
// (16 lanes = one edge row): coalesced float2 gather of X[src], then two
// guaranteed-native GLOBAL_ATOMIC_ADD_F32 (no-return form via inline asm)
// into agg[dst]. X (12.8 MB) and agg (12.8 MB) are both L2-resident
// (192 MB L2), so this is L2 atomic-throughput bound, not HBM bound.
// ---------------------------------------------------------------------------
__device__ __forceinline__ void atomic_add_f32_noret(float* p, float v) {
    asm volatile("global_atomic_add_f32 %0, %1, off"
                 :
                 : "v"(p), "v"(v)
                 : "memory");
}

__global__ __launch_bounds__(256) void gin_scatter_kernel(const float* __restrict__ X,
                                                          const int* __restrict__ src,
                                                          const int* __restrict__ dst,
                                                          float* __restrict__ agg) {
    unsigned tid = blockIdx.x * 256u + threadIdx.x;   // < 25.6M
    unsigned e = tid >> 4;            // edge index (wave-uniform per 16 lanes)
    unsigned d = (tid & 15u) * 2u;    // dim pair
    unsigned s = (unsigned)src[e];
    unsigned t = (unsigned)dst[e];
    const float2 v = *(const float2*)&X[s * 32u + d];
    float* p = &agg[t * 32u + d];
    atomic_add_f32_noret(p,     v.x);
    atomic_add_f32_noret(p + 1, v.y);
}

// ---------------------------------------------------------------------------
// Kernel 3: out = agg @ W via V_WMMA_F32_16X16X4_F32 (fp32 exact vs the f32
// reference). Block = 128 threads = 4 waves; each wave owns one full 16-row
// tile (N_NODES % 16 == 0 -> no per-lane guards anywhere) and both 16-col
// tiles. W staged transposed in LDS so B fragments are contiguous float2
// ds reads. 16 WMMAs per wave.
//
// ISA VGPR layouts (CDNA5 ISA):
//   A 16x4 f32:  lane m=lane&15, VGPR0=K(2*hi), VGPR1=K(2*hi+1), hi=lane>>4
//   B 4x16  f32: lane n=lane&15, VGPR0=K(2*hi), VGPR1=K(2*hi+1)
//   C 16x16 f32: VGPR r -> row (r + 8*hi), col = lane&15
// ---------------------------------------------------------------------------
__global__ __launch_bounds__(128) void gin_gemm_kernel(const float* __restrict__ A,
                                                       const float* __restrict__ W,
                                                       float* __restrict__ out) {
    __shared__ float Wt[32 * 32];   // Wt[n*32 + k] = W[k*32 + n]
    int t = threadIdx.x;
    for (int i = t; i < 1024; i += 128) {
        int k = i >> 5, n = i & 31;
        Wt[n * 32 + k] = W[i];
    }
    __syncthreads();

    int wave = t >> 5;
    int lane = t & 31;
    int m    = lane & 15;
    int hi   = lane >> 4;
    int tile = blockIdx.x * 4 + wave;

    if (tile < N_TILES) {           // uniform per-wave branch; EXEC stays all-1s
        int row0 = tile * 16;
        const float* aptr = A + (size_t)(row0 + m) * 32;

        v2f a[8];
#pragma unroll
        for (int s = 0; s < 8; ++s) {
            int k = 4 * s + 2 * hi;
            a[s].x = aptr[k];
            a[s].y = aptr[k + 1];
        }

#pragma unroll
        for (int nt = 0; nt < 2; ++nt) {
            v8f c = {};
            const float* wp = &Wt[(nt * 16 + m) * 32 + 2 * hi];
#pragma unroll
            for (int s = 0; s < 8; ++s) {
                v2f b;
                b.x = wp[4 * s];
                b.y = wp[4 * s + 1];
                // D = A(16x4) * B(4x16) + C ; fp32 in, fp32 accumulate
                c = __builtin_amdgcn_wmma_f32_16x16x4_f32(
                        /*neg_a=*/false, a[s], /*neg_b=*/false, b,
                        /*c_mod=*/(short)0, c, /*reuse_a=*/false, /*reuse_b=*/false);
            }
            float* op = out + (size_t)(row0 + 8 * hi) * 32 + nt * 16 + m;
#pragma unroll
            for (int r = 0; r < 8; ++r) {
                op[(size_t)r * 32] = c[r];   // unguarded: tile is always full
            }
        }
    }
}

// ---------------------------------------------------------------------------
extern "C" void kernel_launch(void* const* d_in, const int* in_sizes, int n_in,
                              void* d_out, int out_size, void* d_ws, size_t ws_size,
                              hipStream_t stream) {
    const float* X   = (const float*)d_in[0];
    const float* W   = (const float*)d_in[1];
    const int*   src = (const int*)d_in[2];
    const int*   dst = (const int*)d_in[3];
    float*       out = (float*)d_out;
    float*       agg = (float*)d_ws;   // 100000*32 floats = 12.8 MB scratch

    // 1) zero agg (800000 float4's)
    int n4 = (N_NODES * IN_DIM) / 4;
    gin_zero_kernel<<<(n4 + 255) / 256, 256, 0, stream>>>((float4*)agg, n4);

    // 2) scatter-add: 25.6M threads, 2 dims each (100000 blocks exactly)
    gin_scatter_kernel<<<(N_EDGES * 16) / 256, 256, 0, stream>>>(X, src, dst, agg);

    // 3) WMMA GEMM: 4 full 16-row tiles per block
    int gblocks = (N_TILES + 3) / 4;   // 1563
    gin_gemm_kernel<<<gblocks, 128, 0, stream>>>(agg, W, out);
}